// RegL1Loss_46188078301464
// MI455X (gfx1250) — compile-verified
//
#include <hip/hip_runtime.h>

// RegL1Loss on MI455X (gfx1250, wave32).
// Shapes fixed by the reference: B=32, P=64, D=34, L=1<<20.
// Memory-latency-bound gather+reduce; one workgroup per batch image.

typedef __attribute__((ext_vector_type(2))) float v2f;
typedef __attribute__((ext_vector_type(8))) float v8f;

#define THREADS 256
#define P_ 64
#define D_ 34
#define PD_ (P_ * D_)     // 2176
#define L_ 1048576

struct GtEntry { float val; float fidx; float flag; };

__global__ __launch_bounds__(THREADS)
void regl1loss_kernel(const float* __restrict__ preds,
                      const float* __restrict__ gts,
                      float* __restrict__ out)
{
    const int b = blockIdx.x;
    const int t = threadIdx.x;

    const GtEntry* __restrict__ g  = reinterpret_cast<const GtEntry*>(gts) + (size_t)b * PD_;
    const float*   __restrict__ pr = preds + (size_t)b * L_;

    __shared__ int   pflag[P_];
    __shared__ float wsum[8];
    __shared__ float wcnt[8];

    if (t < P_) pflag[t] = 0;
    __syncthreads();

    // Main gather pass: 2176 entries / 256 threads -> 8..9 independent gathers
    // per thread (good MLP; each is a random 4B load into a 4MB preds row).
    float acc = 0.0f;
    for (int i = t; i < PD_; i += THREADS) {
        GtEntry e = g[i];               // contiguous 12B -> b96/b32x3 loads
        if (e.flag > 0.0f) {
            const int idx = (int)e.fidx;            // exact: idx < 2^20 < 2^24
            acc += fabsf(pr[idx] - e.val);
            pflag[i / D_] = 1;          // benign same-value race in LDS
        }
    }
    __syncthreads();                    // all pflag writes visible

    // Person-count partial: threads 0..63 carry one person flag each.
    float cv = (t < P_) ? (float)pflag[t] : 0.0f;

    // ---- Per-wave dual reduction via one V_WMMA_F32_16X16X4_F32 ----
    // A(16x4): lanes 0-15: A[m][0]=acc, A[m][1]=cv ; lanes 16-31: A[m][2]=acc, A[m][3]=cv.
    // B(4x16): B[0][0]=B[2][0]=1 (route acc -> column 0), B[1][1]=B[3][1]=1 (cv -> column 1).
    // => D[m][0] = acc_m + acc_{m+16},  D[m][1] = cv_m + cv_{m+16}, full f32.
    const int laneN = t & 15;
    const int lane  = t & 31;
    const int w     = t >> 5;

    float wave_sum, wave_cnt_or_dead;
#if __has_builtin(__builtin_amdgcn_wmma_f32_16x16x4_f32)
    {
        v2f A;  A[0]  = acc;                      A[1]  = cv;
        v2f Bm; Bm[0] = (laneN == 0) ? 1.f : 0.f; Bm[1] = (laneN == 1) ? 1.f : 0.f;
        v8f C = {};
        v8f Dm = __builtin_amdgcn_wmma_f32_16x16x4_f32(
            /*neg_a=*/false, A, /*neg_b=*/false, Bm,
            /*c_mod=*/(short)0, C, /*reuse_a=*/false, /*reuse_b=*/false);
        // lane n (0-15) holds D[0..7][n]; lane n+16 holds D[8..15][n].
        float r = Dm[0] + Dm[1] + Dm[2] + Dm[3] + Dm[4] + Dm[5] + Dm[6] + Dm[7];
        float r_hi = __shfl_down(r, 16);          // lane0 += lane16, lane1 += lane17
        float tot  = r + r_hi;                    // lane0: wave acc-sum, lane1: wave cv-sum
        wave_sum = tot;                           // meaningful in lane 0
        wave_cnt_or_dead = tot;                   // meaningful in lane 1
    }
#else
    {   // Fallback: log2 shuffle tree (wave32).
        float s = acc, c = cv;
        for (int off = 16; off > 0; off >>= 1) {
            s += __shfl_down(s, off);
            c += __shfl_down(c, off);
        }
        wave_sum = s;                             // lane 0
        wave_cnt_or_dead = __shfl_up(c, 1);       // place count in lane 1
    }
#endif

    if (lane == 0) wsum[w] = wave_sum;
    if (lane == 1) wcnt[w] = wave_cnt_or_dead;
    __syncthreads();

    if (t == 0) {
        float S = 0.f, N = 0.f;
#pragma unroll
        for (int i = 0; i < 8; ++i) { S += wsum[i]; N += wcnt[i]; }
        out[b] = S / N;
    }
}

extern "C" void kernel_launch(void* const* d_in, const int* in_sizes, int n_in,
                              void* d_out, int out_size, void* d_ws, size_t ws_size,
                              hipStream_t stream)
{
    const float* preds = (const float*)d_in[0];   // [B, L] f32
    const float* gts   = (const float*)d_in[1];   // [B, P, D, 3] f32
    float* out = (float*)d_out;                   // [B] f32

    const int B = out_size;                       // 32
    regl1loss_kernel<<<dim3(B), dim3(THREADS), 0, stream>>>(preds, gts, out);
    (void)in_sizes; (void)n_in; (void)d_ws; (void)ws_size;
}